// PhaseFunctionedNetwork_62972810494161
// MI455X (gfx1250) — compile-verified
//
#include <hip/hip_runtime.h>
#include <cstddef>

typedef __attribute__((ext_vector_type(2))) float v2f;
typedef __attribute__((ext_vector_type(8))) float v8f;

namespace {
constexpr int kBatch  = 256;
constexpr int kInFeat = 342;
constexpr int kHid    = 512;
constexpr int kOut    = 311;
constexpr int kK0Pad  = 352;   // input K padded (zero rows) past 342
constexpr int kNSl    = 4;
}

// Kernel 1: per-sample Catmull-Rom slice coefficients + transpose/pad input
// into H0 (K-major: H0[k*256 + b]), zero rows for k >= 342.
__global__ __launch_bounds__(256)
void pfnn_prep_kernel(const float* __restrict__ x, float* __restrict__ coef,
                      float* __restrict__ H0) {
  int idx = blockIdx.x * blockDim.x + threadIdx.x;
  if (idx < kK0Pad * kBatch) {
    int k = idx / kBatch;
    int b = idx - k * kBatch;
    H0[idx] = (k < kInFeat) ? x[(size_t)b * (kInFeat + 1) + k] : 0.0f;
  }
  if (idx < kBatch) {
    float phase = x[(size_t)idx * (kInFeat + 1) + kInFeat];
    float ps  = (float)kNSl * phase;          // in [0, 4)
    float fl  = floorf(ps);
    float mu  = ps - fl;
    int   i1  = ((int)fl) & (kNSl - 1);
    float mu2 = mu * mu, mu3 = mu2 * mu;
    // Catmull-Rom weights for y0..y3 (matches reference cubic())
    float w0 = -0.5f * mu3 +        mu2 - 0.5f * mu;
    float w1 =  1.5f * mu3 - 2.5f * mu2 + 1.0f;
    float w2 = -1.5f * mu3 + 2.0f * mu2 + 0.5f * mu;
    float w3 =  0.5f * mu3 - 0.5f * mu2;
    coef[idx * 4 + ((i1 + 3) & 3)] = w0;   // slice i0 = i1-1 mod 4
    coef[idx * 4 +   i1          ] = w1;   // slice i1
    coef[idx * 4 + ((i1 + 1) & 3)] = w2;   // slice i2
    coef[idx * 4 + ((i1 + 2) & 3)] = w3;   // slice i3
  }
}

// Kernel 2: fused PFNN layer. One wave computes a 16x16 output tile,
// accumulating Y_s = W_s * Hin for all 4 slices with a shared B operand
// (V_WMMA_F32_16X16X4_F32, full f32 precision), then blends per-column:
//   out[m][n] = act( sum_s coef[n][s] * (Y_s[m][n] + bias_s[m]) )
//
// WMMA f32 16x16x4 operand layout (wave32):
//   A 16x4 : v0 = {K0 | K2}, v1 = {K1 | K3} per lane-half, M = lane%16
//   B 4x16 : same striping with N = lane%16
//   C/D    : VGPR r holds M = r + 8*(lane/16), N = lane%16
template <int O, int K, bool ACT, bool TOUT>
__global__ __launch_bounds__(128)
void pfnn_layer_kernel(const float* __restrict__ W, const float* __restrict__ bias,
                       const float* __restrict__ Hin, const float* __restrict__ coef,
                       float* __restrict__ out) {
  constexpr int KMAIN = K & ~3;                 // clamp-free main extent
  constexpr size_t SLICE = (size_t)O * K;       // elements per weight slice
  const int lane = threadIdx.x & 31;
  const int wave = threadIdx.x >> 5;
  const int half = lane >> 4;       // 0: lanes 0-15, 1: lanes 16-31
  const int lhm  = lane & 15;
  const int m0   = blockIdx.x * 16;
  const int n0   = (blockIdx.y * 4 + wave) * 16;
  const int ncol = n0 + lhm;

  // Clamp A row so the final-layer partial tile (O=311) never reads OOB;
  // extra rows are masked at store time.
  const int rowA = (m0 + lhm < O) ? (m0 + lhm) : (O - 1);
  const float* __restrict__ wrow = W + (size_t)rowA * K;
  const float* __restrict__ wr = wrow + 2 * half;            // (k, k+1) pair base
  const float* __restrict__ hp = Hin + (size_t)(2 * half) * kBatch + ncol;

  v8f acc[4] = {};

  for (int k0 = 0; k0 < KMAIN; k0 += 4) {
    v2f bv;
    bv.x = hp[0];
    bv.y = hp[kBatch];
#pragma unroll
    for (int s = 0; s < 4; ++s) {
      const v2f av = *(const v2f*)(wr + (size_t)s * SLICE);  // b64, imm slice offset
      acc[s] = __builtin_amdgcn_wmma_f32_16x16x4_f32(
          false, av, false, bv, (short)0, acc[s], false, false);
    }
    wr += 4;
    hp += 4 * kBatch;
  }

  if constexpr (KMAIN < K) {
    // Tail: k = KMAIN..KMAIN+3. A k-index clamped into [0,K); the clamped
    // lanes multiply zero-padded rows of Hin, contributing 0.
    const int ka = KMAIN + 2 * half;
    const int kb = ka + 1;
    const int kca = (ka < K) ? ka : (K - 1);
    const int kcb = (kb < K) ? kb : (K - 1);
    v2f bv;
    bv.x = hp[0];
    bv.y = hp[kBatch];
#pragma unroll
    for (int s = 0; s < 4; ++s) {
      v2f av;
      av.x = wrow[(size_t)s * SLICE + kca];
      av.y = wrow[(size_t)s * SLICE + kcb];
      acc[s] = __builtin_amdgcn_wmma_f32_16x16x4_f32(
          false, av, false, bv, (short)0, acc[s], false, false);
    }
  }

  // Per-column blend coefficients
  const float c0 = coef[ncol * 4 + 0];
  const float c1 = coef[ncol * 4 + 1];
  const float c2 = coef[ncol * 4 + 2];
  const float c3 = coef[ncol * 4 + 3];

#pragma unroll
  for (int r = 0; r < 8; ++r) {
    const int m = m0 + r + 8 * half;
    const int mc = (m < O) ? m : (O - 1);
    float bb = c0 * bias[0 * O + mc] + c1 * bias[1 * O + mc] +
               c2 * bias[2 * O + mc] + c3 * bias[3 * O + mc];
    float v = c0 * acc[0][r] + c1 * acc[1][r] +
              c2 * acc[2][r] + c3 * acc[3][r] + bb;
    if constexpr (ACT) v = (v > 0.0f) ? v : (expf(v) - 1.0f);  // ELU(alpha=1)
    if (m < O) {
      if constexpr (TOUT) out[(size_t)ncol * O + m] = v;        // (B, OUT)
      else                out[(size_t)m * kBatch + ncol] = v;   // K-major H
    }
  }
}

extern "C" void kernel_launch(void* const* d_in, const int* in_sizes, int n_in,
                              void* d_out, int out_size, void* d_ws, size_t ws_size,
                              hipStream_t stream) {
  (void)in_sizes; (void)n_in; (void)out_size; (void)ws_size;
  const float* x  = (const float*)d_in[0];
  const float* W0 = (const float*)d_in[1];
  const float* b0 = (const float*)d_in[2];
  const float* W1 = (const float*)d_in[3];
  const float* b1 = (const float*)d_in[4];
  const float* W2 = (const float*)d_in[5];
  const float* b2 = (const float*)d_in[6];

  // Workspace (floats): coef[256*4] | H0[352*256] | H1[512*256] | H2[512*256]
  float* ws   = (float*)d_ws;
  float* coef = ws;
  float* H0   = coef + kBatch * 4;
  float* H1   = H0 + kK0Pad * kBatch;
  float* H2   = H1 + kHid * kBatch;

  const int prepN = kK0Pad * kBatch;
  pfnn_prep_kernel<<<(prepN + 255) / 256, 256, 0, stream>>>(x, coef, H0);

  dim3 blk(128);  // 4 waves, one 16x16 tile each; 16 column tiles -> grid.y = 4
  // Layer 0: 342 -> 512, ELU
  pfnn_layer_kernel<kHid, kInFeat, true, false>
      <<<dim3(kHid / 16, 4), blk, 0, stream>>>(W0, b0, H0, coef, H1);
  // Layer 1: 512 -> 512, ELU
  pfnn_layer_kernel<kHid, kHid, true, false>
      <<<dim3(kHid / 16, 4), blk, 0, stream>>>(W1, b1, H1, coef, H2);
  // Layer 2: 512 -> 311, linear, write (B, OUT)
  pfnn_layer_kernel<kOut, kHid, false, true>
      <<<dim3((kOut + 15) / 16, 4), blk, 0, stream>>>(W2, b2, H2, coef, (float*)d_out);
}